// SpectralNet_22016002359686
// MI455X (gfx1250) — compile-verified
//
#include <hip/hip_runtime.h>
#include <math.h>

// ---------------- problem constants (from reference) ----------------
#define NGRID   128
#define PW      4            // window width P
#define LBOX    10.0f
#define CCH     32           // channels
#define RTAP    8            // blur truncation radius (tail ~ e^-32, < fp32 eps)
#define NTAP    (2*RTAP+1)
#define NG3     (NGRID*NGRID*NGRID)
#define PI_F    3.14159265358979323846f

typedef __attribute__((ext_vector_type(2))) float v2f;
typedef __attribute__((ext_vector_type(8))) float v8f;

// ---------------------------------------------------------------
// Kernel 1: per-axis blur taps  g1(d) = (1/NG) * sum_j exp(-0.5*sig^2*k_j^2) * cos(2*pi*j*d/NG)
// ---------------------------------------------------------------
__global__ void taps_kernel(float* __restrict__ taps) {
    int t = threadIdx.x;
    if (t >= NTAP) return;
    const int d = t - RTAP;
    const float h = LBOX / NGRID;
    const float sig = h;
    float acc = 0.f;
    for (int j = 0; j < NGRID; ++j) {
        int jj = (j < NGRID/2) ? j : (j - NGRID);
        float k = 2.f * PI_F * (float)jj / (NGRID * h);
        float g = expf(-0.5f * sig * sig * k * k);
        acc += g * cosf(2.f * PI_F * (float)(j * d) / (float)NGRID);
    }
    taps[t] = acc / (float)NGRID;
}

// ---------------------------------------------------------------
// Kernel 2: MLP (two 32x32 fp32 GEMMs via v_wmma_f32_16x16x4_f32) + channel sum
// One block = 128 threads (4 wave32) = 128 particles.
// LDS rows padded to 33 floats (33 coprime 64 banks -> conflict-free).
// ---------------------------------------------------------------
__global__ __launch_bounds__(128) void mlp_kernel(
    const int*   __restrict__ z,
    const float* __restrict__ emb,
    const float* __restrict__ W0, const float* __restrict__ b0,
    const float* __restrict__ W1, const float* __restrict__ b1,
    float* __restrict__ s_out, int N)
{
    __shared__ float XA[128 * 33];   // input acts / final acts
    __shared__ float XB[128 * 33];   // hidden acts
    __shared__ float WA[32 * 33];    // W0
    __shared__ float WB[32 * 33];    // W1
    __shared__ float BA[32];
    __shared__ float BB[32];

    const int tid  = threadIdx.x;
    const int lane = tid & 31;
    const int wv   = tid >> 5;
    const int n0   = blockIdx.x * 128;

    // stage weights + biases
    #pragma unroll
    for (int k = 0; k < 8; ++k) {
        int idx = tid + k * 128;          // 0..1023
        int r = idx >> 5, c = idx & 31;
        WA[r * 33 + c] = W0[idx];
        WB[r * 33 + c] = W1[idx];
    }
    if (tid < 32) { BA[tid] = b0[tid]; BB[tid] = b1[tid]; }

    // stage embeddings (zero-pad tail rows so WMMA input is defined)
    {
        int n = n0 + tid;
        if (n < N) {
            const float* src = emb + z[n] * CCH;
            #pragma unroll
            for (int c = 0; c < CCH; ++c) XA[tid * 33 + c] = src[c];
        } else {
            #pragma unroll
            for (int c = 0; c < CCH; ++c) XA[tid * 33 + c] = 0.f;
        }
    }
    __syncthreads();

    const int col16  = lane & 15;
    const int khalf  = (lane >> 4) << 1;   // lanes 0-15: K+{0,1}; lanes 16-31: K+{2,3}
    const int rowadd = (lane >> 4) << 3;   // C/D: lanes 16-31 hold rows M+8

    // ---- layer 1: XB = silu(XA @ W0^T + b0) ----
    #pragma unroll
    for (int t = 0; t < 4; ++t) {
        int tile = wv * 4 + t;            // 16 tiles: 8 row-tiles x 2 col-tiles
        int tm = tile >> 1, tn = tile & 1;
        int arow = tm * 16 + col16;
        int bcol = tn * 16 + col16;
        v8f acc = {};
        #pragma unroll
        for (int kk = 0; kk < 8; ++kk) {  // K = 32 in steps of 4
            int klo = kk * 4 + khalf;
            v2f a = { XA[arow * 33 + klo], XA[arow * 33 + klo + 1] };
            v2f b = { WA[bcol * 33 + klo], WA[bcol * 33 + klo + 1] }; // B[k,j]=W0[j,k]
            acc = __builtin_amdgcn_wmma_f32_16x16x4_f32(
                false, a, false, b, (short)0, acc, false, false);
        }
        float bias = BA[bcol];
        #pragma unroll
        for (int r = 0; r < 8; ++r) {
            float v = acc[r] + bias;
            v = v / (1.f + __expf(-v));   // silu
            XB[(tm * 16 + r + rowadd) * 33 + bcol] = v;
        }
    }
    __syncthreads();

    // ---- layer 2: XA = silu(XB @ W1^T + b1) ----
    #pragma unroll
    for (int t = 0; t < 4; ++t) {
        int tile = wv * 4 + t;
        int tm = tile >> 1, tn = tile & 1;
        int arow = tm * 16 + col16;
        int bcol = tn * 16 + col16;
        v8f acc = {};
        #pragma unroll
        for (int kk = 0; kk < 8; ++kk) {
            int klo = kk * 4 + khalf;
            v2f a = { XB[arow * 33 + klo], XB[arow * 33 + klo + 1] };
            v2f b = { WB[bcol * 33 + klo], WB[bcol * 33 + klo + 1] };
            acc = __builtin_amdgcn_wmma_f32_16x16x4_f32(
                false, a, false, b, (short)0, acc, false, false);
        }
        float bias = BB[bcol];
        #pragma unroll
        for (int r = 0; r < 8; ++r) {
            float v = acc[r] + bias;
            v = v / (1.f + __expf(-v));
            XA[(tm * 16 + r + rowadd) * 33 + bcol] = v;
        }
    }
    __syncthreads();

    // ---- channel sum: s[n] = sum_c x[n,c] ----
    {
        int n = n0 + tid;
        if (n < N) {
            float acc = 0.f;
            #pragma unroll
            for (int c = 0; c < CCH; ++c) acc += XA[tid * 33 + c];
            s_out[n] = acc;
        }
    }
}

// ---------------------------------------------------------------
// Kernel 3: scatter  f[cell] += w_n[cell] * s[n]   (one thread per (particle, cell))
// ---------------------------------------------------------------
__global__ __launch_bounds__(256) void scatter_kernel(
    const float* __restrict__ pos, const float* __restrict__ sval,
    float* __restrict__ field, int N)
{
    int g = blockIdx.x * blockDim.x + threadIdx.x;
    int n = g >> 6;          // particle
    int q = g & 63;          // cell within 4x4x4 window
    if (n >= N) return;

    const float h = LBOX / NGRID;
    const float inv_h = (float)NGRID / LBOX;
    const float inv2s2 = 0.5f * inv_h * inv_h;   // 1/(2 sig^2), sig = h

    float px = pos[n * 3 + 0], py = pos[n * 3 + 1], pz = pos[n * 3 + 2];
    px -= floorf(px / LBOX) * LBOX;
    py -= floorf(py / LBOX) * LBOX;
    pz -= floorf(pz / LBOX) * LBOX;
    int bx = (int)floorf(px * inv_h);
    int by = (int)floorf(py * inv_h);
    int bz = (int)floorf(pz * inv_h);

    int a = q >> 4;          // x offset index (slowest in flattened window)
    int b = (q >> 2) & 3;    // y
    int c = q & 3;           // z
    float dx = px - (float)(bx + a - 1) * h;
    float dy = py - (float)(by + b - 1) * h;
    float dz = pz - (float)(bz + c - 1) * h;
    float w = __expf(-(dx * dx + dy * dy + dz * dz) * inv2s2);

    int ix = (bx + a - 1 + NGRID) & (NGRID - 1);
    int iy = (by + b - 1 + NGRID) & (NGRID - 1);
    int iz = (bz + c - 1 + NGRID) & (NGRID - 1);
    int cell = (iz * NGRID + iy) * NGRID + ix;
    atomicAdd(&field[cell], w * sval[n]);
}

// ---------------------------------------------------------------
// Kernel 4: separable periodic blur along one axis (17 taps)
// grid = (128,128), block = 128; one line per block, staged in LDS.
// ---------------------------------------------------------------
__global__ __launch_bounds__(128) void blur_kernel(
    const float* __restrict__ in, float* __restrict__ out,
    const float* __restrict__ taps, int axis)
{
    __shared__ float line[NGRID];
    __shared__ float tp[NTAP];

    int p = threadIdx.x;
    int u = blockIdx.x, v = blockIdx.y;
    if (p < NTAP) tp[p] = taps[p];

    int base, sp;
    if (axis == 0)      { base = (v * NGRID + u) * NGRID; sp = 1; }
    else if (axis == 1) { base = v * NGRID * NGRID + u;   sp = NGRID; }
    else                { base = v * NGRID + u;           sp = NGRID * NGRID; }

    line[p] = in[base + p * sp];
    __syncthreads();

    float acc = 0.f;
    #pragma unroll
    for (int t = 0; t < NTAP; ++t) {
        int d = t - RTAP;
        acc += tp[t] * line[(p - d + NGRID) & (NGRID - 1)];
    }
    out[base + p * sp] = acc;
}

// ---------------------------------------------------------------
// Kernel 5: gather  energy[n] = sum_{64 cells} w_n[cell] * e[cell]
// ---------------------------------------------------------------
__global__ __launch_bounds__(256) void gather_kernel(
    const float* __restrict__ pos, const float* __restrict__ efield,
    float* __restrict__ energy, int N)
{
    int n = blockIdx.x * blockDim.x + threadIdx.x;
    if (n >= N) return;

    const float h = LBOX / NGRID;
    const float inv_h = (float)NGRID / LBOX;
    const float inv2s2 = 0.5f * inv_h * inv_h;

    float px = pos[n * 3 + 0], py = pos[n * 3 + 1], pz = pos[n * 3 + 2];
    px -= floorf(px / LBOX) * LBOX;
    py -= floorf(py / LBOX) * LBOX;
    pz -= floorf(pz / LBOX) * LBOX;
    int bx = (int)floorf(px * inv_h);
    int by = (int)floorf(py * inv_h);
    int bz = (int)floorf(pz * inv_h);

    float wx[PW], wy[PW], wz[PW];
    int   ix[PW], iy[PW], iz[PW];
    #pragma unroll
    for (int t = 0; t < PW; ++t) {
        float dx = px - (float)(bx + t - 1) * h;
        float dy = py - (float)(by + t - 1) * h;
        float dz = pz - (float)(bz + t - 1) * h;
        wx[t] = __expf(-dx * dx * inv2s2);
        wy[t] = __expf(-dy * dy * inv2s2);
        wz[t] = __expf(-dz * dz * inv2s2);
        ix[t] = (bx + t - 1 + NGRID) & (NGRID - 1);
        iy[t] = (by + t - 1 + NGRID) & (NGRID - 1);
        iz[t] = (bz + t - 1 + NGRID) & (NGRID - 1);
    }

    float acc = 0.f;
    #pragma unroll
    for (int a = 0; a < PW; ++a) {
        #pragma unroll
        for (int b = 0; b < PW; ++b) {
            float wxy = wx[a] * wy[b];
            #pragma unroll
            for (int c = 0; c < PW; ++c) {
                acc += wxy * wz[c] * efield[(iz[c] * NGRID + iy[b]) * NGRID + ix[a]];
            }
        }
    }
    energy[n] = acc;
}

// ---------------------------------------------------------------
extern "C" void kernel_launch(void* const* d_in, const int* in_sizes, int n_in,
                              void* d_out, int out_size, void* d_ws, size_t ws_size,
                              hipStream_t stream) {
    const int*   z   = (const int*)  d_in[0];
    const float* pos = (const float*)d_in[1];
    /* d_in[2] = batch: all zeros, NBATCH==1 -> unused */
    const float* emb = (const float*)d_in[3];
    const float* W0  = (const float*)d_in[4];
    const float* b0  = (const float*)d_in[5];
    const float* W1  = (const float*)d_in[6];
    const float* b1  = (const float*)d_in[7];
    float* energy = (float*)d_out;
    const int N = in_sizes[0];

    // workspace layout
    char* ws = (char*)d_ws;
    size_t off = 0;
    float* sval = (float*)(ws + off); off += ((size_t)N * 4 + 255) & ~(size_t)255;
    float* taps = (float*)(ws + off); off += 256;
    float* fieldA = (float*)(ws + off); off += (size_t)NG3 * 4;
    float* fieldB = (float*)(ws + off); off += (size_t)NG3 * 4;
    (void)ws_size; (void)n_in; (void)out_size;

    taps_kernel<<<1, 32, 0, stream>>>(taps);
    mlp_kernel<<<(N + 127) / 128, 128, 0, stream>>>(z, emb, W0, b0, W1, b1, sval, N);
    hipMemsetAsync(fieldA, 0, (size_t)NG3 * 4, stream);
    scatter_kernel<<<(N * 64 + 255) / 256, 256, 0, stream>>>(pos, sval, fieldA, N);
    blur_kernel<<<dim3(NGRID, NGRID), 128, 0, stream>>>(fieldA, fieldB, taps, 0); // x
    blur_kernel<<<dim3(NGRID, NGRID), 128, 0, stream>>>(fieldB, fieldA, taps, 1); // y
    blur_kernel<<<dim3(NGRID, NGRID), 128, 0, stream>>>(fieldA, fieldB, taps, 2); // z
    gather_kernel<<<(N + 255) / 256, 256, 0, stream>>>(pos, fieldB, energy, N);
}